// GraphSAGE_MaxPooling_40218073759863
// MI455X (gfx1250) — compile-verified
//
#include <hip/hip_runtime.h>

#define N_NODES 100000
#define N_EDGES 1600000
#define D 64
#define K2 128   // 2*D

typedef __attribute__((ext_vector_type(2))) float v2f;
typedef __attribute__((ext_vector_type(8))) float v8f;

// ---------------------------------------------------------------------------
// Kernel 1: zero the aggregation workspace. relu'd messages are >= 0, so
// max-accumulating into a 0-initialized buffer reproduces
// maximum(segment_max(msg,u), 0) from the reference exactly.
// ---------------------------------------------------------------------------
__global__ __launch_bounds__(256) void zero_kernel(float4* __restrict__ p, int n4) {
    int i = blockIdx.x * 256 + threadIdx.x;
    if (i < n4) p[i] = make_float4(0.f, 0.f, 0.f, 0.f);
}

// ---------------------------------------------------------------------------
// Kernel 2: edge scatter-max. Each wave32 owns 32 consecutive edges:
// lanes do one coalesced int64 index load each, then per edge the indices
// are broadcast via shuffle and all 32 lanes process 2 feature dims (float2).
// Non-negative floats order identically as their uint bit patterns, so
// atomicMax on the uint view implements float max (traffic stays in L2:
// fts + agg = 51 MB << 192 MB).
// ---------------------------------------------------------------------------
__global__ __launch_bounds__(256) void scatter_kernel(const float* __restrict__ fts,
                                                      const long long* __restrict__ ei,
                                                      float* __restrict__ agg) {
    const int lane      = threadIdx.x & 31;
    const int wave_glb  = (blockIdx.x * 256 + threadIdx.x) >> 5;
    const int edge_base = wave_glb * 32;           // N_EDGES % 32 == 0
    if (edge_base >= N_EDGES) return;

    // Coalesced index loads: lane e fetches edge (edge_base + e)'s endpoints.
    const int my_dst = (int)ei[edge_base + lane];            // u: destination
    const int my_src = (int)ei[N_EDGES + edge_base + lane];  // v: source

    const int d = lane * 2;
#pragma unroll 4
    for (int e = 0; e < 32; ++e) {
        const int dst = __shfl(my_dst, e, 32);
        const int src = __shfl(my_src, e, 32);

        float2 x = *(const float2*)(fts + (long long)src * D + d);
        float a = fmaxf(x.x, 0.f);
        float b = fmaxf(x.y, 0.f);

        unsigned* p = (unsigned*)(agg + (long long)dst * D + d);
        if (a > 0.f) atomicMax(p,     __float_as_uint(a));
        if (b > 0.f) atomicMax(p + 1, __float_as_uint(b));
    }
}

// ---------------------------------------------------------------------------
// Kernel 3: fused GEMM (concat[fts,agg] @ W_l via V_WMMA_F32_16X16X4_F32)
// + row L2 normalization. 4 waves/block, 16 nodes per wave, N=64 covered by
// 4 accumulator tiles, K=128 in 32 WMMA steps of K=4 (split 16 fts / 16 agg).
// W is staged in LDS pre-swizzled so a B fragment {W[k][j], W[k+1][j]} is a
// single aligned ds_load_b64.
// ---------------------------------------------------------------------------
__global__ __launch_bounds__(128) void gemm_norm_kernel(const float* __restrict__ fts,
                                                        const float* __restrict__ agg,
                                                        const float* __restrict__ W,
                                                        float* __restrict__ out) {
    // w_swz[p][2*j + b] = W[2*p + b][j]  (p = k/2, b = k&1) -> 32 KB
    __shared__ float w_swz[(K2 / 2) * (2 * D)];
    __shared__ float o_lds[4][16 * 65];   // per-wave 16x64 tile, pad 65

    const int tid  = threadIdx.x;
    const int wave = tid >> 5;
    const int lane = tid & 31;

    // Stage + swizzle W_l [128 x 64]: coalesced float4 read, strided scatter.
    for (int i = tid; i < (K2 * D) / 4; i += 128) {
        const int k  = i >> 4;            // row of W (16 float4 per row)
        const int j0 = (i & 15) * 4;      // starting column
        const float4 w4 = ((const float4*)W)[i];
        float* dstp = &w_swz[(k >> 1) * (2 * D) + (k & 1)];
        dstp[2 * (j0 + 0)] = w4.x;
        dstp[2 * (j0 + 1)] = w4.y;
        dstp[2 * (j0 + 2)] = w4.z;
        dstp[2 * (j0 + 3)] = w4.w;
    }
    __syncthreads();

    const int m  = lane & 15;             // row (A) / column (B,C,D) in tile
    const int kh = (lane >> 4) << 1;      // K sub-offset: 0 or 2
    const int node_base = blockIdx.x * 64 + wave * 16;

    int node = node_base + m;
    if (node > N_NODES - 1) node = N_NODES - 1;   // clamp reads; stores guarded
    const float* frow = fts + (long long)node * D;
    const float* arow = agg + (long long)node * D;

    v8f acc[4] = {v8f{}, v8f{}, v8f{}, v8f{}};

    // First half of K: columns 0..63 of concat come from fts.
#pragma unroll 4
    for (int kk = 0; kk < 16; ++kk) {
        const int k = kk * 4 + kh;
        const float2 a2 = *(const float2*)(frow + k);
        v2f A; A.x = a2.x; A.y = a2.y;
        const float* bp = &w_swz[(k >> 1) * (2 * D)];
#pragma unroll
        for (int t = 0; t < 4; ++t) {
            v2f B = *(const v2f*)(bp + 2 * (t * 16 + m));   // ds_load_b64
            acc[t] = __builtin_amdgcn_wmma_f32_16x16x4_f32(
                false, A, false, B, (short)0, acc[t], false, false);
        }
    }
    // Second half of K: columns 64..127 come from agg.
#pragma unroll 4
    for (int kk = 16; kk < 32; ++kk) {
        const int k = kk * 4 + kh;
        const float2 a2 = *(const float2*)(arow + (k - D));
        v2f A; A.x = a2.x; A.y = a2.y;
        const float* bp = &w_swz[(k >> 1) * (2 * D)];
#pragma unroll
        for (int t = 0; t < 4; ++t) {
            v2f B = *(const v2f*)(bp + 2 * (t * 16 + m));
            acc[t] = __builtin_amdgcn_wmma_f32_16x16x4_f32(
                false, A, false, B, (short)0, acc[t], false, false);
        }
    }

    // Spill C/D tiles: VGPR r, lanes 0-15 -> row r; lanes 16-31 -> row r+8;
    // column = lane%16 (+ 16*t).
    const int rhalf = lane >> 4;
#pragma unroll
    for (int t = 0; t < 4; ++t) {
#pragma unroll
        for (int r = 0; r < 8; ++r) {
            o_lds[wave][(r + 8 * rhalf) * 65 + t * 16 + m] = acc[t][r];
        }
    }
    __syncthreads();

    // Row L2 norm: lane handles row = lane%16 over half the 64 cols; the two
    // halves combine with a lane-16 xor shuffle.
    const int row   = lane & 15;
    const int chalf = (lane >> 4) * 32;
    float ss = 0.f;
#pragma unroll
    for (int c = 0; c < 32; ++c) {
        const float v = o_lds[wave][row * 65 + chalf + c];
        ss += v * v;
    }
    ss += __shfl_xor(ss, 16, 32);
    const float rn = 1.0f / sqrtf(ss);   // matches reference (no eps)

    const int onode = node_base + row;
    if (onode < N_NODES) {
        float* orow = out + (long long)onode * D + chalf;
#pragma unroll
        for (int c = 0; c < 32; c += 4) {
            float4 v4;
            v4.x = o_lds[wave][row * 65 + chalf + c + 0] * rn;
            v4.y = o_lds[wave][row * 65 + chalf + c + 1] * rn;
            v4.z = o_lds[wave][row * 65 + chalf + c + 2] * rn;
            v4.w = o_lds[wave][row * 65 + chalf + c + 3] * rn;
            *(float4*)(orow + c) = v4;
        }
    }
}

// ---------------------------------------------------------------------------
extern "C" void kernel_launch(void* const* d_in, const int* in_sizes, int n_in,
                              void* d_out, int out_size, void* d_ws, size_t ws_size,
                              hipStream_t stream) {
    const float*     fts = (const float*)d_in[0];
    const long long* ei  = (const long long*)d_in[1];  // int64 edge_index [2, E]
    const float*     W   = (const float*)d_in[2];      // [128, 64] row-major
    float*           out = (float*)d_out;
    float*           agg = (float*)d_ws;               // [N_NODES, 64] scratch

    // 1) zero agg
    const int n4 = (N_NODES * D) / 4;
    zero_kernel<<<(n4 + 255) / 256, 256, 0, stream>>>((float4*)agg, n4);

    // 2) scatter-max over edges (32 edges per wave, coalesced index loads)
    const int swaves  = N_EDGES / 32;                  // 50,000 waves
    const int sblocks = (swaves * 32 + 255) / 256;     // 6,250 blocks
    scatter_kernel<<<sblocks, 256, 0, stream>>>(fts, ei, agg);

    // 3) fused WMMA GEMM + L2 normalize
    const int gblocks = (N_NODES + 63) / 64;
    gemm_norm_kernel<<<gblocks, 128, 0, stream>>>(fts, agg, W, out);
}